// GaussianRenderer_90537910599855
// MI455X (gfx1250) — compile-verified
//
#include <hip/hip_runtime.h>
#include <math.h>

// Problem constants (match reference)
#define RW  800
#define RH  600
#define RHW (RW * RH)

typedef float v2f  __attribute__((ext_vector_type(2)));
typedef float v8f  __attribute__((ext_vector_type(8)));
typedef unsigned int v4u __attribute__((ext_vector_type(4)));
typedef int   v4i  __attribute__((ext_vector_type(4)));
typedef int   v8i_t __attribute__((ext_vector_type(8)));

// ---------------------------------------------------------------------------
// Kernel 0: tiny prep — invert 4x4 extrinsics, compose P = K @ Tinv[:3,:]
// mats[0..11] = P (3x4 row major), mats[12..15] = Tinv row 2 (camera z)
// ---------------------------------------------------------------------------
__global__ void gr_prep(const float* __restrict__ K, const float* __restrict__ E,
                        float* __restrict__ mats) {
    if (threadIdx.x != 0 || blockIdx.x != 0) return;
    float A[4][8];
    for (int i = 0; i < 4; ++i)
        for (int j = 0; j < 4; ++j) {
            A[i][j]     = E[i * 4 + j];
            A[i][4 + j] = (i == j) ? 1.0f : 0.0f;
        }
    // Gauss-Jordan with partial pivoting
    for (int col = 0; col < 4; ++col) {
        int piv = col;
        for (int r = col + 1; r < 4; ++r)
            if (fabsf(A[r][col]) > fabsf(A[piv][col])) piv = r;
        if (piv != col)
            for (int j = 0; j < 8; ++j) { float t = A[col][j]; A[col][j] = A[piv][j]; A[piv][j] = t; }
        float inv = 1.0f / A[col][col];
        for (int j = 0; j < 8; ++j) A[col][j] *= inv;
        for (int r = 0; r < 4; ++r) {
            if (r == col) continue;
            float f = A[r][col];
            for (int j = 0; j < 8; ++j) A[r][j] -= f * A[col][j];
        }
    }
    for (int n = 0; n < 3; ++n)
        for (int k = 0; k < 4; ++k) {
            float s = 0.0f;
            for (int j = 0; j < 3; ++j) s += K[n * 3 + j] * A[j][4 + k];
            mats[n * 4 + k] = s;
        }
    for (int k = 0; k < 4; ++k) mats[12 + k] = A[2][4 + k];
}

// ---------------------------------------------------------------------------
// Kernel 1: projection via V_WMMA_F32_16X16X4_F32.
// Each wave32 projects 16 points: A(16x4) = [x y z 1] per point, B(4x16)
// columns 0..3 = {P row0, P row1, P row2, Tinv row2}. D gives
// [sx sy sw camz] per point. D is unscrambled through LDS per the ISA
// 16x16 f32 C/D layout (VGPR r, lanes0-15: M=r; lanes16-31: M=r+8).
// Emits per-point record {xi, yi, depth, opacity, r, g, b, valid} and a
// 64-bit sort key (depth-descending order, stable by index).
// ---------------------------------------------------------------------------
__global__ __launch_bounds__(256) void gr_project(
    const float* __restrict__ xyz, const float* __restrict__ opacity,
    const float* __restrict__ features, const float* __restrict__ mats,
    float* __restrict__ rec, unsigned long long* __restrict__ keys,
    int N, int NP) {
    __shared__ float xch[8][32][8];   // 8 waves x 32 lanes x 8 D-VGPRs = 8KB

    const int lane = threadIdx.x & 31;
    const int wave = threadIdx.x >> 5;
    const int base = (blockIdx.x * 8 + wave) * 16;

    // ---- A matrix (16x4 f32): lanes0-15 v0=K0(x) v1=K1(y); lanes16-31 v0=K2(z) v1=K3(1)
    const int mA = lane & 15;
    const int pA = base + mA;
    const int cp = (pA < N) ? pA : (N - 1);
    const float msk = (pA < N) ? 1.0f : 0.0f;
    const float px = xyz[3 * cp + 0] * msk;
    const float py = xyz[3 * cp + 1] * msk;
    const float pz = xyz[3 * cp + 2] * msk;
    v2f a;
    a.x = (lane < 16) ? px : pz;
    a.y = (lane < 16) ? py : 1.0f;

    // ---- B matrix (4x16 f32): lanes0-15 v0=K0 v1=K1; lanes16-31 v0=K2 v1=K3
    const int nB = lane & 15;
    const int k0 = (lane & 16) ? 2 : 0;
    const int idx0 = (nB < 3) ? (nB * 4 + k0) : (12 + k0);
    const float bmsk = (nB < 4) ? 1.0f : 0.0f;
    v2f b;
    b.x = mats[idx0]     * bmsk;
    b.y = mats[idx0 + 1] * bmsk;

    v8f c = {};
    c = __builtin_amdgcn_wmma_f32_16x16x4_f32(false, a, false, b,
                                              (short)0, c, false, false);

    #pragma unroll
    for (int r = 0; r < 8; ++r) xch[wave][lane][r] = c[r];
    __syncthreads();

    if (lane < 16) {
        const int m  = lane;
        const int n0 = (m >= 8) ? 16 : 0;
        const int r  = m & 7;
        const float sx = xch[wave][n0 + 0][r];
        const float sy = xch[wave][n0 + 1][r];
        const float sw = xch[wave][n0 + 2][r];
        const float cz = xch[wave][n0 + 3][r];

        const int p = base + m;
        const float fpx = sx / sw;
        const float fpy = sy / sw;
        const bool valid = (p < N) && (fpx >= 0.0f) && (fpx < (float)RW) &&
                           (fpy >= 0.0f) && (fpy < (float)RH) && (cz > 0.0f);
        const int xi = (int)fpx;   // truncation toward zero, fpx >= 0 when valid
        const int yi = (int)fpy;

        const int cq = (p < N) ? p : (N - 1);
        const float op = opacity[cq];
        const float cr = features[cq * 48 + 0];
        const float cg = features[cq * 48 + 1];
        const float cb = features[cq * 48 + 2];

        float* rp = rec + (size_t)p * 8;
        rp[0] = valid ? (float)xi : -1.0e9f;
        rp[1] = valid ? (float)yi : -1.0e9f;
        rp[2] = cz;
        rp[3] = op;
        rp[4] = cr;
        rp[5] = cg;
        rp[6] = cb;
        rp[7] = valid ? 1.0f : 0.0f;

        // Sort key: ascending u64 == descending depth; invalid/pad sorts last.
        const unsigned u = __float_as_uint(cz);
        const unsigned msel = (u >> 31) ? 0xFFFFFFFFu : 0x80000000u;
        const unsigned k32 = valid ? ~(u ^ msel) : 0xFFFFFFFFu;
        keys[p] = ((unsigned long long)k32 << 32) | (unsigned)p;
    }
}

// ---------------------------------------------------------------------------
// Kernel 2: one bitonic compare-exchange pass (ascending u64 sort)
// ---------------------------------------------------------------------------
__global__ __launch_bounds__(256) void gr_bitonic(unsigned long long* __restrict__ keys,
                                                  unsigned j, unsigned k) {
    const unsigned i = blockIdx.x * 256u + threadIdx.x;
    const unsigned l = i ^ j;
    if (l > i) {
        const unsigned long long va = keys[i];
        const unsigned long long vb = keys[l];
        const bool asc = ((i & k) == 0u);
        const bool sw  = asc ? (va > vb) : (va < vb);
        if (sw) { keys[i] = vb; keys[l] = va; }
    }
}

// ---------------------------------------------------------------------------
// Kernel 3: gather records into sorted order (contiguous stream for blending)
// ---------------------------------------------------------------------------
__global__ __launch_bounds__(256) void gr_gather(const unsigned long long* __restrict__ keys,
                                                 const float* __restrict__ rec,
                                                 float* __restrict__ srec, int NP) {
    const int i = blockIdx.x * 256 + threadIdx.x;
    if (i < NP) {
        const unsigned idx = (unsigned)(keys[i] & 0xFFFFFFFFull);
        const float4* s = (const float4*)(rec + (size_t)idx * 8);
        float4* d = (float4*)(srec + (size_t)i * 8);
        d[0] = s[0];
        d[1] = s[1];
    }
}

// ---------------------------------------------------------------------------
// Kernel 4: per-tile ordered blend. One 16x16 pixel tile per block, one pixel
// per thread. Streams the sorted point list in 256-point chunks through LDS
// (Tensor Data Mover when available), votes per-chunk on tile intersection,
// and composites back-to-front in registers. 4MB point list lives in L2.
// ---------------------------------------------------------------------------
__global__ __launch_bounds__(256) void gr_blend(const float* __restrict__ srec,
                                                float* __restrict__ out, int NP) {
    __shared__ float4 smem[512];   // 256 records x 32B = 8KB

    const int tx = threadIdx.x & 15;
    const int ty = threadIdx.x >> 4;
    const int X  = blockIdx.x * 16 + tx;
    const int Y  = blockIdx.y * 16 + ty;
    const float fx = (float)X;
    const float fy = (float)Y;
    const float bx0 = (float)((int)blockIdx.x * 16 - 3);
    const float bx1 = (float)((int)blockIdx.x * 16 + 18);
    const float by0 = (float)((int)blockIdx.y * 16 - 3);
    const float by1 = (float)((int)blockIdx.y * 16 + 18);

    float rr = 0.0f, gg = 0.0f, bb = 0.0f, al = 0.0f, dm = 0.0f;

    const int nchunks = NP / 256;
    for (int c = 0; c < nchunks; ++c) {
        __syncthreads();   // previous chunk fully consumed before overwrite
        const float* gsrc = srec + (size_t)c * 2048;   // 8KB chunk
        if (c + 1 < nchunks) {
            // hint next chunk toward L2/L0 (global_prefetch_b8)
            __builtin_prefetch((const char*)(gsrc + 2048) + threadIdx.x * 32, 0, 1);
        }
#if __has_builtin(__builtin_amdgcn_tensor_load_to_lds) && __has_builtin(__builtin_amdgcn_s_wait_tensorcnt)
        if (threadIdx.x < 32) {   // one wave drives the TDM
            const unsigned long long ga = (unsigned long long)(const void*)gsrc;
            const unsigned ldsoff = (unsigned)(unsigned long long)(void*)smem;
            v4u g0;
            g0.x = 1u;                                   // count=1, user descriptor
            g0.y = ldsoff;                               // lds_addr (bytes)
            g0.z = (unsigned)(ga & 0xFFFFFFFFull);       // global_addr[31:0]
            g0.w = (unsigned)((ga >> 32) & 0x01FFFFFFull) | (2u << 30); // addr[56:32] | type=2
            v8i_t g1;
            g1[0] = 0x20000;                  // workgroup_mask=0, data_size=2 (4B)
            g1[1] = (int)(2048u << 16);       // tensor_dim0 = 2048 (low 16 in [31:16])
            g1[2] = (int)(1u << 16);          // tensor_dim0 hi=0, tensor_dim1 = 1
            g1[3] = (int)(2048u << 16);       // tensor_dim1 hi=0, tile_dim0 = 2048
            g1[4] = 1;                        // tile_dim1 = 1, tile_dim2 = 0
            g1[5] = 2048;                     // tensor_dim0_stride low
            g1[6] = 0;
            g1[7] = 0;
            v4i  g2 = {0, 0, 0, 0};
            v4i  g3 = {0, 0, 0, 0};
            v8i_t g4 = {0, 0, 0, 0, 0, 0, 0, 0};   // 6-arg toolchain form
            __builtin_amdgcn_tensor_load_to_lds(g0, g1, g2, g3, g4, 0);
            __builtin_amdgcn_s_wait_tensorcnt(0);
        }
        __syncthreads();
#else
        {   // fallback: cooperative vector load through VGPRs into LDS
            const float4* g4 = (const float4*)gsrc;
            smem[threadIdx.x * 2 + 0] = g4[threadIdx.x * 2 + 0];
            smem[threadIdx.x * 2 + 1] = g4[threadIdx.x * 2 + 1];
        }
        __syncthreads();
#endif
        // per-chunk votes: any valid point left? any point touching this tile?
        const float4 mh = smem[threadIdx.x * 2 + 0];
        const float4 mt = smem[threadIdx.x * 2 + 1];
        const int myvalid = (mt.w != 0.0f) ? 1 : 0;
        const int myinter = myvalid &&
                            (mh.x >= bx0) && (mh.x <= bx1) &&
                            (mh.y >= by0) && (mh.y <= by1);
        const int anyvalid = __syncthreads_or(myvalid);
        const int anyinter = __syncthreads_or(myinter);
        if (!anyvalid) break;          // sorted: everything after is padding/invalid
        if (!anyinter) continue;       // chunk misses this tile entirely

        for (int j = 0; j < 256; ++j) {
            const float4 h = smem[2 * j + 0];   // {xi, yi, depth, opacity}
            const float4 t = smem[2 * j + 1];   // {r, g, b, valid}
            const float dx = fx - h.x;
            const float dy = fy - h.y;
            if ((t.w != 0.0f) & (dx >= -3.0f) & (dx <= 3.0f) &
                (dy >= -3.0f) & (dy <= 3.0f)) {
                const float aa = h.w * expf(-0.5f * (dx * dx + dy * dy));
                const float ia = 1.0f - aa;
                rr = rr * ia + t.x * aa;
                gg = gg * ia + t.y * aa;
                bb = bb * ia + t.z * aa;
                al = al + aa * (1.0f - al);
                dm = (dm == 0.0f || h.z < dm) ? h.z : dm;
            }
        }
    }

    if (X < RW && Y < RH) {
        const int p = Y * RW + X;
        out[p]            = rr;
        out[RHW + p]      = gg;
        out[2 * RHW + p]  = bb;
        out[3 * RHW + p]  = dm;   // depth map
        out[4 * RHW + p]  = al;   // alpha map
    }
}

// ---------------------------------------------------------------------------
// Host launcher. Inputs (setup_inputs order): xyz, rotation, scaling, opacity,
// features, camera_intrinsics, camera_extrinsics. rotation/scaling are unused
// by the reference. Output: image(3,H,W) ++ depth(1,H,W) ++ alpha(1,H,W).
// Workspace (NP = 2^ceil(log2 N) = 131072): rec NP*32B | keys NP*8B |
// srec NP*32B | mats 256B  -> ~9.5MB.
// ---------------------------------------------------------------------------
extern "C" void kernel_launch(void* const* d_in, const int* in_sizes, int n_in,
                              void* d_out, int out_size, void* d_ws, size_t ws_size,
                              hipStream_t stream) {
    const float* xyz      = (const float*)d_in[0];
    const float* opacity  = (const float*)d_in[3];
    const float* features = (const float*)d_in[4];
    const float* Km       = (const float*)d_in[5];
    const float* Em       = (const float*)d_in[6];

    const int N = in_sizes[0] / 3;
    int NP = 256;
    while (NP < N) NP <<= 1;

    char* ws = (char*)d_ws;
    float*              rec  = (float*)ws;
    unsigned long long* keys = (unsigned long long*)(ws + (size_t)NP * 32);
    float*              srec = (float*)(ws + (size_t)NP * 40);
    float*              mats = (float*)(ws + (size_t)NP * 72);

    gr_prep<<<1, 32, 0, stream>>>(Km, Em, mats);
    gr_project<<<NP / 128, 256, 0, stream>>>(xyz, opacity, features, mats,
                                             rec, keys, N, NP);
    for (unsigned k = 2; k <= (unsigned)NP; k <<= 1)
        for (unsigned j = k >> 1; j >= 1; j >>= 1)
            gr_bitonic<<<NP / 256, 256, 0, stream>>>(keys, j, k);
    gr_gather<<<NP / 256, 256, 0, stream>>>(keys, rec, srec, NP);

    dim3 tiles((RW + 15) / 16, (RH + 15) / 16);
    gr_blend<<<tiles, 256, 0, stream>>>(srec, (float*)d_out, NP);
}